// EdgeGatedGraphConv_2594160247294
// MI455X (gfx1250) — compile-verified
//
#include <hip/hip_runtime.h>
#include <math.h>

#define NN 100000
#define NE 1000000
#define D 64
#define ROWSTRIDE 68   // LDS row stride (dwords) for C-layout spill: 16B-aligned, conflict-free
#define WPITCH 65      // LDS row stride (dwords) for staged W: bank-conflict-free B-frag reads

typedef float v2f __attribute__((ext_vector_type(2)));
typedef float v8f __attribute__((ext_vector_type(8)));

__device__ __forceinline__ v8f wmma4(v2f a, v2f b, v8f c) {
  // D = A(16x4 f32) * B(4x16 f32) + C(16x16 f32)
  return __builtin_amdgcn_wmma_f32_16x16x4_f32(false, a, false, b, (short)0, c,
                                               false, false);
}

// acc[t] (t=0..3 -> output cols 16t..16t+15) = X[x0..x0+15][0..63] @ W[64][64] + bias
// A layout: lane l: M = l%16, K = 2*(l/16) + j   (j = VGPR index)
// B layout: lane l: N = l%16, row K = 2*(l/16) + j
// C layout: lane l: N = l%16, row M = v + 8*(l/16) (v = VGPR index 0..7)
__device__ __forceinline__ void gemm16x64(const float* __restrict__ X, int x0,
                                          const float* __restrict__ W,
                                          const float* __restrict__ bias,
                                          int lm, int half, v8f acc[4]) {
#pragma unroll
  for (int t = 0; t < 4; ++t) {
    float bv = bias[t * 16 + lm];
    acc[t] = (v8f){bv, bv, bv, bv, bv, bv, bv, bv};
  }
  const float* xrow = X + (size_t)(x0 + lm) * D;
#pragma unroll
  for (int k = 0; k < 16; ++k) {
    int kk = 4 * k + 2 * half;
    v2f a = *(const v2f*)(xrow + kk);
#pragma unroll
    for (int t = 0; t < 4; ++t) {
      v2f b;
      b.x = W[kk * D + t * 16 + lm];
      b.y = W[(kk + 1) * D + t * 16 + lm];
      acc[t] = wmma4(a, b, acc[t]);
    }
  }
}

// Same GEMM, but B-fragments come from W staged in LDS with row pitch WPITCH.
__device__ __forceinline__ void gemm16x64_ldsW(const float* __restrict__ X, int x0,
                                               const float* Wlds,
                                               const float* __restrict__ bias,
                                               int lm, int half, v8f acc[4]) {
#pragma unroll
  for (int t = 0; t < 4; ++t) {
    float bv = bias[t * 16 + lm];
    acc[t] = (v8f){bv, bv, bv, bv, bv, bv, bv, bv};
  }
  const float* xrow = X + (size_t)(x0 + lm) * D;
#pragma unroll
  for (int k = 0; k < 16; ++k) {
    int kk = 4 * k + 2 * half;
    v2f a = *(const v2f*)(xrow + kk);
#pragma unroll
    for (int t = 0; t < 4; ++t) {
      v2f b;
      b.x = Wlds[kk * WPITCH + t * 16 + lm];        // rows kk / kk+1, 65-dword apart
      b.y = Wlds[(kk + 1) * WPITCH + t * 16 + lm];  // -> ds_load_2addr candidates
      acc[t] = wmma4(a, b, acc[t]);
    }
  }
}

__device__ __forceinline__ float sigm(float x) {
  return 1.0f / (1.0f + __expf(-x));
}

// ---------------- Kernel 1: node projections (4 GEMMs) + zero accumulators ----
__global__ void node_proj_kernel(const float* __restrict__ nf,
                                 const float* __restrict__ Wsg, const float* __restrict__ bsg,
                                 const float* __restrict__ Wdg, const float* __restrict__ bdg,
                                 const float* __restrict__ Wdu, const float* __restrict__ bdu,
                                 const float* __restrict__ Wsu, const float* __restrict__ bsu,
                                 float* __restrict__ e_src, float* __restrict__ e_dst,
                                 float* __restrict__ Bh, float* __restrict__ Ah,
                                 float* __restrict__ Ssh, float* __restrict__ Ss) {
  int lane = threadIdx.x & 31;
  int wid  = threadIdx.x >> 5;     // 4 waves: one weight matrix each
  int lm = lane & 15, half = lane >> 4;
  int n0 = blockIdx.x * 16;        // 100000 / 16 = 6250 exact tiles

  const float* W = (wid == 0) ? Wsg : (wid == 1) ? Wdg : (wid == 2) ? Wdu : Wsu;
  const float* b = (wid == 0) ? bsg : (wid == 1) ? bdg : (wid == 2) ? bdu : bsu;
  float* out     = (wid == 0) ? e_src : (wid == 1) ? e_dst : (wid == 2) ? Bh : Ah;

  v8f acc[4];
  gemm16x64(nf, n0, W, b, lm, half, acc);
#pragma unroll
  for (int t = 0; t < 4; ++t)
#pragma unroll
    for (int v = 0; v < 8; ++v)
      out[(size_t)(n0 + v + 8 * half) * D + t * 16 + lm] = acc[t][v];

  // zero segment-sum accumulators for this node tile (16*64 floats * 2 arrays)
  float4 z = {0.f, 0.f, 0.f, 0.f};
  int base4 = n0 * (D / 4);
  for (int i = threadIdx.x; i < 16 * (D / 4); i += 128) {
    ((float4*)Ssh)[base4 + i] = z;
    ((float4*)Ss)[base4 + i]  = z;
  }
}

// ---------------- Kernel 2: fused edge pipeline ------------------------------
__global__ void edge_kernel(const float* __restrict__ ef,
                            const int* __restrict__ src, const int* __restrict__ dst,
                            const float* __restrict__ Weg, const float* __restrict__ beg,
                            const float* __restrict__ e_src, const float* __restrict__ e_dst,
                            const float* __restrict__ Bh,
                            const float* __restrict__ gamma_e, const float* __restrict__ beta_e,
                            float* __restrict__ Ssh, float* __restrict__ Ss,
                            float* __restrict__ edge_out) {
  __shared__ float Wlds[D * WPITCH];         // 16.6 KB: W_edge_gate, padded rows
  __shared__ float buf[4][16 * ROWSTRIDE];   // 17.4 KB: per-wave 16x64 proj spill
  int lane = threadIdx.x & 31;
  int wid  = threadIdx.x >> 5;
  int lm = lane & 15, half = lane >> 4;
  int e0 = (blockIdx.x * 4 + wid) * 16;      // 1e6 / 16 = 62500 exact tiles

  // Stage W_edge_gate -> LDS via async global->LDS copies (ASYNCcnt-tracked).
  // Each lane supplies its own LDS destination, inserting the WPITCH padding.
#pragma unroll
  for (int i = threadIdx.x; i < D * 16; i += 128) {   // 64 rows * 16 quads of 16B
    int r = i >> 4, q = (i & 15) << 2;
    const float* g = Weg + r * D + q;
    unsigned l = (unsigned)(uintptr_t)(&Wlds[r * WPITCH + q]);
    asm volatile("global_load_async_to_lds_b128 %0, %1, off"
                 :: "v"(l), "v"(g) : "memory");
  }
  asm volatile("s_wait_asynccnt 0x0" ::: "memory");
  __syncthreads();

  // edge gate projection: ef_tile @ W_edge_gate + b  (fp32 WMMA, B-frags from LDS)
  v8f acc[4];
  gemm16x64_ldsW(ef, e0, Wlds, beg, lm, half, acc);

  float* mybuf = buf[wid];
#pragma unroll
  for (int t = 0; t < 4; ++t)
#pragma unroll
    for (int v = 0; v < 8; ++v)
      mybuf[(v + 8 * half) * ROWSTRIDE + t * 16 + lm] = acc[t][v];

  __syncthreads();   // cross-lane handoff: WMMA C-layout -> row layout

  // row layout: lanes (l, l+16) own edge lm; halves of the 64 features
  int m = lm, F = half * 32;
  int e = e0 + m;
  int s = src[e], d = dst[e];
  const float* esr = e_src + (size_t)s * D + F;
  const float* edr = e_dst + (size_t)d * D + F;
  const float* bhr = Bh + (size_t)s * D + F;
  const float* efr = ef + (size_t)e * D + F;   // cache-hot (just streamed by GEMM)
  float* sshp = Ssh + (size_t)d * D + F;
  float* ssp  = Ss + (size_t)d * D + F;

  float4 ge[8];
  float sum = 0.f, ssq = 0.f;
#pragma unroll
  for (int c4 = 0; c4 < 8; ++c4) {
    int f = c4 * 4;
    float4 p  = *(const float4*)(&mybuf[m * ROWSTRIDE + F + f]);
    float4 ga = *(const float4*)(esr + f);
    float4 gb = *(const float4*)(edr + f);
    float4 xf = *(const float4*)(efr + f);
    float4 bh = *(const float4*)(bhr + f);
    float4 sg;
    sg.x = sigm(p.x + ga.x + gb.x);
    sg.y = sigm(p.y + ga.y + gb.y);
    sg.z = sigm(p.z + ga.z + gb.z);
    sg.w = sigm(p.w + ga.w + gb.w);
    // segment sums (reference: segment_sum(Bh[src]*sigma, dst), segment_sum(sigma, dst))
    unsafeAtomicAdd(sshp + f + 0, bh.x * sg.x);
    unsafeAtomicAdd(sshp + f + 1, bh.y * sg.y);
    unsafeAtomicAdd(sshp + f + 2, bh.z * sg.z);
    unsafeAtomicAdd(sshp + f + 3, bh.w * sg.w);
    unsafeAtomicAdd(ssp + f + 0, sg.x);
    unsafeAtomicAdd(ssp + f + 1, sg.y);
    unsafeAtomicAdd(ssp + f + 2, sg.z);
    unsafeAtomicAdd(ssp + f + 3, sg.w);
    float4 g;
    g.x = sg.x * xf.x; g.y = sg.y * xf.y; g.z = sg.z * xf.z; g.w = sg.w * xf.w;
    ge[c4] = g;
    sum += g.x + g.y + g.z + g.w;
    ssq += g.x * g.x + g.y * g.y + g.z * g.z + g.w * g.w;
  }
  // combine the two feature-halves of each edge row (wave32 lane-pair)
  sum += __shfl_xor(sum, 16);
  ssq += __shfl_xor(ssq, 16);
  float mu   = sum * (1.f / 64.f);
  float var  = fmaxf(ssq * (1.f / 64.f) - mu * mu, 0.f);
  float rstd = rsqrtf(var + 1e-5f);

  float* outp = edge_out + (size_t)e * D + F;
#pragma unroll
  for (int c4 = 0; c4 < 8; ++c4) {
    int f = c4 * 4;
    float4 gm = *(const float4*)(gamma_e + F + f);
    float4 bt = *(const float4*)(beta_e + F + f);
    float4 g = ge[c4];
    float4 o;
    o.x = (g.x - mu) * rstd * gm.x + bt.x;
    o.y = (g.y - mu) * rstd * gm.y + bt.y;
    o.z = (g.z - mu) * rstd * gm.z + bt.z;
    o.w = (g.w - mu) * rstd * gm.w + bt.w;
    *(float4*)(outp + f) = o;
  }
}

// ---------------- Kernel 3: node output (h = Ssh/(Ss+eps); LN(Ah + h)) -------
__global__ void node_out_kernel(const float* __restrict__ Ah,
                                const float* __restrict__ Ssh, const float* __restrict__ Ss,
                                const float* __restrict__ gamma_n, const float* __restrict__ beta_n,
                                float* __restrict__ node_out) {
  int lane = threadIdx.x & 31;
  int wid  = threadIdx.x >> 5;
  int lm = lane & 15, half = lane >> 4;
  int n = blockIdx.x * 64 + wid * 16 + lm;
  int F = half * 32;
  bool ok = (n < NN);

  float4 xs[8];
  float sum = 0.f, ssq = 0.f;
  if (ok) {
    const float* ah = Ah + (size_t)n * D + F;
    const float* sh = Ssh + (size_t)n * D + F;
    const float* sr = Ss + (size_t)n * D + F;
#pragma unroll
    for (int c4 = 0; c4 < 8; ++c4) {
      int f = c4 * 4;
      float4 a  = *(const float4*)(ah + f);
      float4 h4 = *(const float4*)(sh + f);
      float4 s4 = *(const float4*)(sr + f);
      float4 x;
      x.x = a.x + h4.x / (s4.x + 1e-8f);
      x.y = a.y + h4.y / (s4.y + 1e-8f);
      x.z = a.z + h4.z / (s4.z + 1e-8f);
      x.w = a.w + h4.w / (s4.w + 1e-8f);
      xs[c4] = x;
      sum += x.x + x.y + x.z + x.w;
      ssq += x.x * x.x + x.y * x.y + x.z * x.z + x.w * x.w;
    }
  }
  sum += __shfl_xor(sum, 16);
  ssq += __shfl_xor(ssq, 16);
  if (ok) {
    float mu   = sum * (1.f / 64.f);
    float var  = fmaxf(ssq * (1.f / 64.f) - mu * mu, 0.f);
    float rstd = rsqrtf(var + 1e-5f);
    float* op = node_out + (size_t)n * D + F;
#pragma unroll
    for (int c4 = 0; c4 < 8; ++c4) {
      int f = c4 * 4;
      float4 gm = *(const float4*)(gamma_n + F + f);
      float4 bt = *(const float4*)(beta_n + F + f);
      float4 x = xs[c4];
      float4 o;
      o.x = (x.x - mu) * rstd * gm.x + bt.x;
      o.y = (x.y - mu) * rstd * gm.y + bt.y;
      o.z = (x.z - mu) * rstd * gm.z + bt.z;
      o.w = (x.w - mu) * rstd * gm.w + bt.w;
      *(float4*)(op + f) = o;
    }
  }
}

extern "C" void kernel_launch(void* const* d_in, const int* in_sizes, int n_in,
                              void* d_out, int out_size, void* d_ws, size_t ws_size,
                              hipStream_t stream) {
  (void)in_sizes; (void)n_in; (void)out_size; (void)ws_size;
  const float* nf  = (const float*)d_in[0];
  const float* efe = (const float*)d_in[1];
  const int*   src = (const int*)d_in[2];
  const int*   dst = (const int*)d_in[3];
  const float* Wsg = (const float*)d_in[4];  const float* bsg = (const float*)d_in[5];
  const float* Wdg = (const float*)d_in[6];  const float* bdg = (const float*)d_in[7];
  const float* Weg = (const float*)d_in[8];  const float* beg = (const float*)d_in[9];
  const float* Wsu = (const float*)d_in[10]; const float* bsu = (const float*)d_in[11];
  const float* Wdu = (const float*)d_in[12]; const float* bdu = (const float*)d_in[13];
  const float* g_e = (const float*)d_in[14]; const float* b_e = (const float*)d_in[15];
  const float* g_n = (const float*)d_in[16]; const float* b_n = (const float*)d_in[17];

  size_t nd = (size_t)NN * D;
  float* ws    = (float*)d_ws;
  float* e_src = ws;
  float* e_dst = e_src + nd;
  float* Bh    = e_dst + nd;
  float* Ah    = Bh + nd;
  float* Ssh   = Ah + nd;
  float* Ss    = Ssh + nd;

  float* node_out = (float*)d_out;            // [NN*D] then [NE*D]
  float* edge_out = node_out + nd;

  node_proj_kernel<<<NN / 16, 128, 0, stream>>>(nf, Wsg, bsg, Wdg, bdg, Wdu, bdu,
                                                Wsu, bsu, e_src, e_dst, Bh, Ah, Ssh, Ss);
  edge_kernel<<<NE / 64, 128, 0, stream>>>(efe, src, dst, Weg, beg, e_src, e_dst, Bh,
                                           g_e, b_e, Ssh, Ss, edge_out);
  node_out_kernel<<<(NN + 63) / 64, 128, 0, stream>>>(Ah, Ssh, Ss, g_n, b_n, node_out);
}